// model_18975165513872
// MI455X (gfx1250) — compile-verified
//
#include <hip/hip_runtime.h>
#include <stdint.h>
#include <float.h>

typedef unsigned long long u64;
typedef unsigned int u32;
typedef __attribute__((ext_vector_type(2))) float v2f;
typedef __attribute__((ext_vector_type(8))) float v8f;

#define K_TOP   100
#define NDET    1000
#define NCAT    80
#define NBATCH  4
#define NLAYER  5
#define CHUNK   32768
#define NTHREADS 256
#define MAXBLK  64   // candidate stride per (side,layer,batch)

// ---- order-preserving float<->u32 and 64-bit sort keys -------------------
// key = ord(score)<<32 | (0xFFFFFFFF - idx)  => descending u64 sort gives
// (score desc, idx asc), matching jax lax.top_k tie-breaking exactly.
__device__ __forceinline__ u32 ordf(float f) {
    u32 u = __float_as_uint(f);
    return (u & 0x80000000u) ? ~u : (u | 0x80000000u);
}
__device__ __forceinline__ float unordf(u32 o) {
    u32 u = (o & 0x80000000u) ? (o & 0x7FFFFFFFu) : ~o;
    return __uint_as_float(u);
}
__device__ __forceinline__ u64 mkkey(float f, u32 idx) {
    return ((u64)ordf(f) << 32) | (u64)(0xFFFFFFFFu - idx);
}
__device__ __forceinline__ u32 keyidx(u64 k) { return 0xFFFFFFFFu - (u32)(k & 0xFFFFFFFFu); }
__device__ __forceinline__ float keyval(u64 k) { return unordf((u32)(k >> 32)); }

// ---- descending bitonic sort of n (power of 2) u64 keys in LDS -----------
__device__ void bitonic_desc(u64* a, int n, int tid, int nthr) {
    for (int k = 2; k <= n; k <<= 1) {
        for (int j = k >> 1; j > 0; j >>= 1) {
            __syncthreads();
            for (int i = tid; i < n; i += nthr) {
                int ixj = i ^ j;
                if (ixj > i) {
                    u64 x = a[i], y = a[ixj];
                    bool up = ((i & k) == 0);          // descending overall
                    if (up ? (x < y) : (x > y)) { a[i] = y; a[ixj] = x; }
                }
            }
        }
    }
    __syncthreads();
}

// ===========================================================================
// Stage 1: fused 3x3 max-pool NMS + per-chunk top-100 (threshold-pruned)
// grid = (nblk, batch, side)
// ===========================================================================
__global__ __launch_bounds__(NTHREADS)
void stage1_nms_topk(const float* __restrict__ tl_heat, const float* __restrict__ br_heat,
                     int H, int W, u64* __restrict__ cand, int layer) {
    __shared__ u64 keys[512];
    __shared__ int cnt;
    __shared__ float thr;
    const int tid  = threadIdx.x;
    const int b    = blockIdx.y;
    const int side = blockIdx.z;
    const int HW = H * W;
    const int total = NCAT * HW;
    const float* heat = (side == 0 ? tl_heat : br_heat) + (size_t)b * total;
    const int start = blockIdx.x * CHUNK;
    const int end = min(start + CHUNK, total);

    for (int q = tid; q < 512; q += NTHREADS) keys[q] = 0ull;
    if (tid == 0) { cnt = 0; thr = 0.0f; }
    __syncthreads();

    for (int base = start; base < end; base += NTHREADS) {
        int g = base + tid;
        bool live = (g < end);
        int gg = live ? g : start;                 // keep loads in-range, branchless
        int c = gg / HW, r = gg - c * HW;
        int y = r / W, x = r - y * W;
        const float* plane = heat + (size_t)c * HW;
        float v = plane[r];
        float m = v;
        #pragma unroll
        for (int dy = -1; dy <= 1; ++dy) {
            #pragma unroll
            for (int dx = -1; dx <= 1; ++dx) {
                int yy = y + dy, xx = x + dx;
                bool inb = (yy >= 0) & (yy < H) & (xx >= 0) & (xx < W);
                int yyc = min(max(yy, 0), H - 1);
                int xxc = min(max(xx, 0), W - 1);
                float nv = plane[yyc * W + xxc];   // unconditional (L1 hit), select after
                m = fmaxf(m, inb ? nv : -FLT_MAX);
            }
        }
        float kept = (v == m) ? v : 0.0f;          // suppressed -> 0 (never in top-100)
        if (live && kept > thr) {
            int pos = atomicAdd(&cnt, 1);
            if (pos < 412) keys[100 + pos] = mkkey(kept, (u32)g);
        }
        __syncthreads();
        if (cnt > 156) {                           // flush before buffer can overflow
            bitonic_desc(keys, 512, tid, NTHREADS);
            if (tid == 0) { thr = keyval(keys[99]); cnt = 0; }
            for (int q = 100 + tid; q < 512; q += NTHREADS) keys[q] = 0ull;
            __syncthreads();
        }
    }
    bitonic_desc(keys, 512, tid, NTHREADS);
    u64* out = cand + ((size_t)((side * NLAYER + layer) * NBATCH + b) * MAXBLK + blockIdx.x) * K_TOP;
    for (int q = tid; q < K_TOP; q += NTHREADS) out[q] = keys[q];
}

// ===========================================================================
// Stage 2: merge chunk candidates -> exact global top-100 per (side,layer,b)
// grid = (layer, batch, side); dyn LDS = 8192*8B
// ===========================================================================
__device__ const int g_dim[NLAYER] = {160, 80, 40, 20, 10};

__global__ __launch_bounds__(NTHREADS)
void stage2_merge(const u64* __restrict__ cand, u64* __restrict__ topk) {
    extern __shared__ u64 skeys[];
    const int tid = threadIdx.x;
    const int layer = blockIdx.x, b = blockIdx.y, side = blockIdx.z;
    const int d = g_dim[layer];
    const int total = NCAT * d * d;
    const int nblk = (total + CHUNK - 1) / CHUNK;
    const int nc = nblk * K_TOP;
    const u64* in = cand + (size_t)((side * NLAYER + layer) * NBATCH + b) * MAXBLK * K_TOP;
    for (int i = tid; i < 8192; i += NTHREADS) skeys[i] = (i < nc) ? in[i] : 0ull;
    bitonic_desc(skeys, 8192, tid, NTHREADS);
    u64* out = topk + (size_t)((side * NLAYER + layer) * NBATCH + b) * K_TOP;
    for (int i = tid; i < K_TOP; i += NTHREADS) out[i] = skeys[i];
}

// ===========================================================================
// Stage 3: pairwise decode. Outer-sum tl_s[i]+br_s[j] via V_WMMA_F32_16X16X4_F32
//   A[m][:] = [tl_s[m], 1, 0, 0],  B[:][n] = [1, br_s[n], 0, 0]  => D = outer sum
// grid = (batch); one launch per layer; dyn LDS = 128KB keys + corner arrays
// ===========================================================================
#define PAD_SLOT 16383   // dummy key slot: out-of-range elements store sentinel 0 here

__global__ __launch_bounds__(NTHREADS)
void stage3_pairs(const u64* __restrict__ topk,
                  const float* __restrict__ tl_regr, const float* __restrict__ br_regr,
                  const float* __restrict__ tl_cregr, const float* __restrict__ br_cregr,
                  float* __restrict__ dets, int layer, int H, int W, float scale) {
    extern __shared__ char smem[];
    u64* keys = (u64*)smem;                     // 16384 keys (131072 B)
    float* fb = (float*)(smem + 16384 * sizeof(u64));
    float* tlx = fb;            float* tly = fb + 112;  float* tls_ = fb + 224;
    float* tlcx = fb + 336;     float* tlcy = fb + 448;
    float* brx = fb + 560;      float* bry = fb + 672;  float* brs_ = fb + 784;
    float* brcx = fb + 896;     float* brcy = fb + 1008;
    int* tlcl = (int*)(fb + 1120);
    int* brcl = tlcl + 112;

    const int tid = threadIdx.x;
    const int b = blockIdx.x;
    const int HW = H * W;

    for (int i = tid; i < 16384; i += NTHREADS) keys[i] = 0ull;
    if (tid < 112) { tls_[tid] = 0.0f; brs_[tid] = 0.0f; }   // WMMA padding rows/cols
    __syncthreads();

    // ---- phase A: decode top-100 corners, gather regressions ----
    if (tid < 2 * K_TOP) {
        int side = tid / K_TOP, k = tid - side * K_TOP;
        u64 key = topk[(size_t)((side * NLAYER + layer) * NBATCH + b) * K_TOP + k];
        float s = keyval(key);
        u32 ind = keyidx(key);
        int cls = (int)(ind / (u32)HW);
        int rem = (int)(ind - (u32)cls * (u32)HW);
        int y = rem / W, x = rem - y * W;
        const float* regr  = (side == 0 ? tl_regr  : br_regr)  + (size_t)b * 2 * HW;
        const float* cregr = (side == 0 ? tl_cregr : br_cregr) + (size_t)b * 2 * HW;
        float rx = regr[rem], ry = regr[HW + rem];
        float cx = cregr[rem], cy = cregr[HW + rem];
        if (side == 0) { tlx[k] = (float)x + rx; tly[k] = (float)y + ry; tls_[k] = s;
                         tlcx[k] = cx; tlcy[k] = cy; tlcl[k] = cls; }
        else           { brx[k] = (float)x + rx; bry[k] = (float)y + ry; brs_[k] = s;
                         brcx[k] = cx; brcy[k] = cy; brcl[k] = cls; }
    }
    __syncthreads();

    // ---- phase B: 7x7 tiles of 16x16 outer sums + mask logic ----
    const float BMAX = (float)(1.3 * 96.0);
    const int lane = tid & 31, wave = tid >> 5;
    const int half = lane >> 4, l16 = lane & 15;
    for (int t = wave; t < 49; t += 8) {        // wave-uniform: EXEC all-ones at WMMA
        int ti = t / 7, tj = t - (t / 7) * 7;
        int j = tj * 16 + l16;
        // unconditional LDS reads (padded/clamped) -> v_cndmask, no exec branches
        float av = tls_[ti * 16 + l16];          // padded to 0 for rows >= 100
        float bv = brs_[j];                      // padded to 0 for cols >= 100
        float sums[8];
#if defined(__gfx1250__) && __has_builtin(__builtin_amdgcn_wmma_f32_16x16x4_f32)
        v2f a, bm;
        a.x  = half ? 0.0f : av;                 // A: K=0 (lanes 0-15) / K=2 (16-31)
        a.y  = half ? 0.0f : 1.0f;               //    K=1 / K=3
        bm.x = half ? 0.0f : 1.0f;               // B: row 0 / row 2
        bm.y = half ? 0.0f : bv;                 //    row 1 / row 3
        v8f c = {};
        v8f dacc = __builtin_amdgcn_wmma_f32_16x16x4_f32(false, a, false, bm,
                                                         (short)0, c, false, false);
        #pragma unroll
        for (int v = 0; v < 8; ++v) sums[v] = dacc[v];
#else
        #pragma unroll
        for (int v = 0; v < 8; ++v) {
            int ii = min(ti * 16 + v + (half ? 8 : 0), 111);
            sums[v] = tls_[ii] + bv;
        }
#endif
        // per-column (j) corner data: hoisted out of the v-loop, clamped loads
        int jc = min(j, K_TOP - 1);
        float bxv = brx[jc], byv = bry[jc];
        float bcx = brcx[jc], bcy = brcy[jc];
        int bcl = brcl[jc];
        bool jok = (j < K_TOP);
        #pragma unroll
        for (int v = 0; v < 8; ++v) {
            int i = ti * 16 + v + (half ? 8 : 0);   // C/D layout: M = v (+8 hi half)
            int ic = min(i, K_TOP - 1);
            float txv = tlx[ic], tyv = tly[ic];
            float tcx = tlcx[ic], tcy = tlcy[ic];
            int tcl = tlcl[ic];
            float bw = bxv - txv;
            float bh = byv - tyv;
            float dsx = fabsf(1.0f - 10.0f * (tcx + bcx) / bw);
            float dsy = fabsf(1.0f - 10.0f * (tcy + bcy) / bh);
            float dd = fabsf(bcy - tcy) + fabsf(bcx - tcx);
            bool bad = (tcl != bcl) ||
                       (bw < 0.0f) || (bw > BMAX) || (bh < 0.0f) || (bh > BMAX) ||
                       (dsx > 0.2f) || (dsy > 0.2f) || (dd > 0.25f);
            float sc = bad ? -1.0f : 0.5f * sums[v];
            int p = i * K_TOP + j;
            bool ok = jok && (i < K_TOP);
            // unconditional store: invalid elements write sentinel 0 into the pad
            // slot (pad region must stay 0 anyway) -> no exec-mask branching
            keys[ok ? p : PAD_SLOT] = ok ? mkkey(sc, (u32)p) : 0ull;
        }
    }
    __syncthreads();

    // ---- phase C/D: top-1000 of 10000 and emit detection rows ----
    bitonic_desc(keys, 16384, tid, NTHREADS);
    for (int s = tid; s < NDET; s += NTHREADS) {
        u64 key = keys[s];
        u32 p = keyidx(key);
        int i = (int)(p / K_TOP), j = (int)(p - (u32)i * K_TOP);
        float* o = dets + ((size_t)b * (NLAYER * NDET) + (size_t)layer * NDET + s) * 8;
        o[0] = tlx[i] * scale; o[1] = tly[i] * scale;
        o[2] = brx[j] * scale; o[3] = bry[j] * scale;
        o[4] = keyval(key);    o[5] = tls_[i]; o[6] = brs_[j]; o[7] = (float)tlcl[i];
    }
}

// ===========================================================================
// Stage 4: full sort of 5000 detections per batch by (score desc, pos asc)
// grid = (batch); dyn LDS = 8192*8B
// ===========================================================================
__global__ __launch_bounds__(NTHREADS)
void stage4_final(const float* __restrict__ dets, float* __restrict__ out) {
    extern __shared__ u64 skeys[];
    const int tid = threadIdx.x, b = blockIdx.x;
    const int N = NLAYER * NDET;
    for (int i = tid; i < 8192; i += NTHREADS)
        skeys[i] = (i < N) ? mkkey(dets[((size_t)b * N + i) * 8 + 4], (u32)i) : 0ull;
    bitonic_desc(skeys, 8192, tid, NTHREADS);
    for (int s = tid; s < N; s += NTHREADS) {
        u32 p = keyidx(skeys[s]);
        const float* src = dets + ((size_t)b * N + p) * 8;
        float* dst = out + ((size_t)b * N + s) * 8;
        #pragma unroll
        for (int f = 0; f < 8; ++f) dst[f] = src[f];
    }
}

// ===========================================================================
extern "C" void kernel_launch(void* const* d_in, const int* in_sizes, int n_in,
                              void* d_out, int out_size, void* d_ws, size_t ws_size,
                              hipStream_t stream) {
    (void)in_sizes; (void)n_in; (void)out_size; (void)ws_size;
    static const int dim[NLAYER] = {160, 80, 40, 20, 10};

    // workspace layout (u64-aligned): candidates | topk | dets
    u64* cand = (u64*)d_ws;                              // 2*5*4*64*100 u64 = 2.048 MB
    u64* topk = cand + (size_t)2 * NLAYER * NBATCH * MAXBLK * K_TOP;
    float* dets = (float*)(topk + (size_t)2 * NLAYER * NBATCH * K_TOP);  // 4*5000*8 f32

    // inputs in setup_inputs() dict order: per layer {tl_heat, br_heat, tl_regr,
    // br_regr, tl_cregr, br_cregr}
    for (int l = 0; l < NLAYER; ++l) {
        int total = NCAT * dim[l] * dim[l];
        int nblk = (total + CHUNK - 1) / CHUNK;
        stage1_nms_topk<<<dim3(nblk, NBATCH, 2), NTHREADS, 0, stream>>>(
            (const float*)d_in[6 * l + 0], (const float*)d_in[6 * l + 1],
            dim[l], dim[l], cand, l);
    }
    stage2_merge<<<dim3(NLAYER, NBATCH, 2), NTHREADS, 8192 * sizeof(u64), stream>>>(cand, topk);
    for (int l = 0; l < NLAYER; ++l) {
        size_t sh = 16384 * sizeof(u64) + (12 * 112) * sizeof(float);
        float scale = 160.0f / (float)dim[l];
        stage3_pairs<<<dim3(NBATCH), NTHREADS, sh, stream>>>(
            topk,
            (const float*)d_in[6 * l + 2], (const float*)d_in[6 * l + 3],
            (const float*)d_in[6 * l + 4], (const float*)d_in[6 * l + 5],
            dets, l, dim[l], dim[l], scale);
    }
    stage4_final<<<dim3(NBATCH), NTHREADS, 8192 * sizeof(u64), stream>>>(dets, (float*)d_out);
}